// QParam_26456998543465
// MI455X (gfx1250) — compile-verified
//
#include <hip/hip_runtime.h>

#define QMAXF 31.0f

typedef __attribute__((ext_vector_type(2))) float v2f;
typedef __attribute__((ext_vector_type(8))) float v8f;

// Tournament merge keeping the lowest index on ties (matches argmin's first-hit rule
// as long as the left operand always carries the lower index range).
#define MERGE(dA, iA, dB, iB)                                   \
    do {                                                        \
        float _db = (dB); int _ib = (iB);                       \
        bool _cc = _db < (dA);                                  \
        (dA) = _cc ? _db : (dA);                                \
        (iA) = _cc ? _ib : (iA);                                \
    } while (0)

// ---------------- Pass 0: init workspace scalar ----------------
__global__ void amax_init_kernel(unsigned* ws) { ws[0] = 0u; }

// ---------------- Pass 1: absolute-max reduction (b128 loads) ----------------
// max(|max(x,0)|, |min(x,0)|) == max|x|, so a single amax suffices.
__global__ void amax_reduce_kernel(const float* __restrict__ x,
                                   unsigned* __restrict__ ws, int n) {
    int tid = blockIdx.x * blockDim.x + threadIdx.x;
    int stride = gridDim.x * blockDim.x;
    float m = 0.0f;
    int nv = n >> 2;
    const float4* x4 = (const float4*)x;
    for (int i = tid; i < nv; i += stride) {
        float4 v = x4[i];
        m = fmaxf(m, fmaxf(fmaxf(fabsf(v.x), fabsf(v.y)),
                           fmaxf(fabsf(v.z), fabsf(v.w))));
    }
    for (int i = (nv << 2) + tid; i < n; i += stride) m = fmaxf(m, fabsf(x[i]));
#pragma unroll
    for (int off = 16; off > 0; off >>= 1)
        m = fmaxf(m, __shfl_xor(m, off, 32));
    // all values >= 0 -> IEEE bit order == value order
    if ((threadIdx.x & 31) == 0) atomicMax(ws, __float_as_uint(m));
}

// ---------------- Pass 2: quantize + codebook argmin via WMMA ----------------
// Score tile: D[m,n] = p_m^2 - 2*p_m*q_n (== |q_n - p_m|^2 - q_n^2), computed as
// A(16x4) x B(4x16) with A[m]=(-2p_m, p_m^2, 0, 0), B[n]=(q_n, 1, 0, 0), C=0.
// Rows = codebook (8 tiles of 16), columns = elements: each column's 16 row-scores
// live in one lane (8 VGPRs) plus its partner lane (lane^16).
__global__ void __launch_bounds__(256)
quant_codebook_kernel(const float* __restrict__ x,
                      const float* __restrict__ plist,
                      const unsigned* __restrict__ ws,
                      float* __restrict__ out, int n) {
    __shared__ float sp[128];
    const int tid = threadIdx.x;
    if (tid < 128) sp[tid] = plist[tid];
    __syncthreads();

    const int lane = tid & 31;
    const bool lo = lane < 16;
    const int hi8 = lo ? 0 : 8;

    const float amax = __uint_as_float(ws[0]);
    const float scale = amax * (1.0f / QMAXF);
    const float inv = QMAXF / amax; // q = x * inv

    // Codebook A operands, resident for the whole kernel (16 VGPRs).
    v2f A[8];
#pragma unroll
    for (int t = 0; t < 8; ++t) {
        float p = lo ? sp[t * 16 + lane] : 0.0f;
        A[t].x = -2.0f * p;
        A[t].y = p * p;
    }
    const float b1 = lo ? 1.0f : 0.0f;
    const v8f cz = {};

    const int wave = (blockIdx.x * blockDim.x + tid) >> 5;
    const int nwaves = (gridDim.x * blockDim.x) >> 5;
    const long step = (long)nwaves * 128; // 128 elements per wave per iteration

    for (long base = (long)wave * 128; base < n; base += step) {
        const long gi0 = base + (long)lane * 4;
        const bool full = (base + 128 <= (long)n); // wave-uniform

        float xv[4];
        if (full) {
            float4 t4 = *(const float4*)(x + gi0); // global_load_b128
            xv[0] = t4.x; xv[1] = t4.y; xv[2] = t4.z; xv[3] = t4.w;
            if (base + step + 128 <= (long)n)
                __builtin_prefetch(x + base + step + (long)lane * 4, 0, 1);
        } else {
#pragma unroll
            for (int j = 0; j < 4; ++j)
                xv[j] = (gi0 + j < (long)n) ? x[gi0 + j] : 0.0f;
        }

        float q[4], qs[4];
#pragma unroll
        for (int j = 0; j < 4; ++j) {
            float t = xv[j] * inv;
            t = fminf(fmaxf(t, -QMAXF), QMAXF);
            q[j] = t;
            qs[j] = __shfl_xor(t, 16, 32); // other half's elements
        }

        float res[4][2];
#pragma unroll
        for (int j = 0; j < 4; ++j) {
#pragma unroll
            for (int h = 0; h < 2; ++h) {
                v2f B;
                B.x = lo ? (h ? qs[j] : q[j]) : 0.0f;
                B.y = b1;

                float dt[8]; int it[8];
#pragma unroll
                for (int t = 0; t < 8; ++t) {
                    v8f D = __builtin_amdgcn_wmma_f32_16x16x4_f32(
                        false, A[t], false, B, (short)0, cz, false, false);
                    // balanced in-tile tournament (independent chains -> ILP)
                    float d0 = D[0], d1 = D[2], d2 = D[4], d3 = D[6];
                    int i0 = t * 16 + 0, i1 = t * 16 + 2,
                        i2 = t * 16 + 4, i3 = t * 16 + 6;
                    MERGE(d0, i0, D[1], t * 16 + 1);
                    MERGE(d1, i1, D[3], t * 16 + 3);
                    MERGE(d2, i2, D[5], t * 16 + 5);
                    MERGE(d3, i3, D[7], t * 16 + 7);
                    MERGE(d0, i0, d1, i1);
                    MERGE(d2, i2, d3, i3);
                    MERGE(d0, i0, d2, i2);
                    dt[t] = d0; it[t] = i0;
                }
                // cross-tile tournament
                MERGE(dt[0], it[0], dt[1], it[1]);
                MERGE(dt[2], it[2], dt[3], it[3]);
                MERGE(dt[4], it[4], dt[5], it[5]);
                MERGE(dt[6], it[6], dt[7], it[7]);
                MERGE(dt[0], it[0], dt[2], it[2]);
                MERGE(dt[4], it[4], dt[6], it[6]);
                MERGE(dt[0], it[0], dt[4], it[4]);

                float bd = dt[0];
                int idx = it[0] + hi8; // full codebook index before the half-swap
                float od = __shfl_xor(bd, 16, 32);
                int oi = __shfl_xor(idx, 16, 32);
                bool take = (od < bd) || (od == bd && oi < idx); // argmin tie rule
                idx = take ? oi : idx;
                res[j][h] = sp[idx];
            }
        }

        float4 o4;
        o4.x = scale * (lo ? res[0][0] : res[0][1]);
        o4.y = scale * (lo ? res[1][0] : res[1][1]);
        o4.z = scale * (lo ? res[2][0] : res[2][1]);
        o4.w = scale * (lo ? res[3][0] : res[3][1]);

        if (full) {
            *(float4*)(out + gi0) = o4; // global_store_b128
        } else {
            float ov[4] = {o4.x, o4.y, o4.z, o4.w};
#pragma unroll
            for (int j = 0; j < 4; ++j)
                if (gi0 + j < (long)n) out[gi0 + j] = ov[j];
        }
    }
}

extern "C" void kernel_launch(void* const* d_in, const int* in_sizes, int n_in,
                              void* d_out, int out_size, void* d_ws, size_t ws_size,
                              hipStream_t stream) {
    (void)n_in; (void)out_size; (void)ws_size;
    const float* x = (const float*)d_in[0];
    const float* plist = (const float*)d_in[1];
    float* out = (float*)d_out;
    unsigned* ws = (unsigned*)d_ws;
    const int n = in_sizes[0]; // 2*1024*1024

    amax_init_kernel<<<1, 1, 0, stream>>>(ws);
    amax_reduce_kernel<<<1024, 256, 0, stream>>>(x, ws, n);
    // 2048 blocks * 8 waves * 128 elem = 2,097,152 elements -> single sweep
    quant_codebook_kernel<<<2048, 256, 0, stream>>>(x, plist, ws, out, n);
}